// LogRatio_14104672600201
// MI455X (gfx1250) — compile-verified
//
#include <hip/hip_runtime.h>
#include <hip/hip_bf16.h>
#include <math.h>

// MI455X / gfx1250, wave32. Fused LogRatio loss:
//   sim tile via V_WMMA_F32_16X16X4_F32 (exact fp32), fused log/mask/row-reduce.
// Phase 1: 16x16 sim tiles per wave, 9 row accumulators, shfl_xor reduce, atomics.
// Phase 2: per-row combine + global sum.

typedef __attribute__((ext_vector_type(2))) float v2f;
typedef __attribute__((ext_vector_type(8))) float v8f;

static constexpr int kN = 4096;
static constexpr int kD = 128;
static constexpr float kEPS = 1e-6f;

__global__ __launch_bounds__(256) void lr_phase1(
    const float* __restrict__ x, const long long* __restrict__ labels,
    float* __restrict__ agg, float a0, float a1, float a2, float a3)
{
  const int lane = threadIdx.x & 31;
  const int wave = threadIdx.x >> 5;
  const int wi = wave >> 1;              // 0..3  -> i sub-tile
  const int wj = wave & 1;               // 0..1  -> j sub-tile
  const int ibase = blockIdx.x * 64 + wi * 16;
  const int jcol0 = blockIdx.y * 512;
  const int hf   = lane >> 4;            // lane half: 0 or 1
  const int l15  = lane & 15;
  const int koff = hf * 2;               // A/B per-lane K offset (16x4 f32 layout)

  // Preload A operand for all K: lane holds rows M=l15, K = {k+koff, k+koff+1}, k=0,4,...,124
  v2f Areg[32];
  {
    const float* ap = x + (ibase + l15) * kD + koff;
#pragma unroll
    for (int kk = 0; kk < 32; ++kk)
      Areg[kk] = *(const v2f*)(ap + kk * 4);
  }

  // targets for this lane-half's 8 rows: C/D VGPR r -> row mi = r + 8*hf
  int trow[8];
#pragma unroll
  for (int r = 0; r < 8; ++r)
    trow[r] = (int)labels[(ibase + r + 8 * hf) * 4 + 0];

  const float a0q = a0 * a0, a1q = a1 * a1, a2q = a2 * a2, a3q = a3 * a3;

  // per-row accumulators: c,S1,S2,P0,P1,P2,Q0,Q1,R0
  float accq[8][9];
#pragma unroll
  for (int r = 0; r < 8; ++r)
#pragma unroll
    for (int q = 0; q < 9; ++q) accq[r][q] = 0.f;

  for (int jt = 0; jt < 16; ++jt) {
    const int jbase = jcol0 + jt * 32 + wj * 16;
    const int j = jbase + l15;           // this lane's column (same for all r)

    // 16x16 sim tile: 32 x wmma_f32_16x16x4_f32 over K=128
    v8f acc = {};
    const float* bp = x + j * kD + koff;
#pragma unroll
    for (int kk = 0; kk < 32; ++kk) {
      v2f b = *(const v2f*)(bp + kk * 4);
      acc = __builtin_amdgcn_wmma_f32_16x16x4_f32(
          /*neg_a=*/false, Areg[kk], /*neg_b=*/false, b,
          /*c_mod=*/(short)0, acc, /*reuse_a=*/false, /*reuse_b=*/false);
    }

    const int lj0 = (int)labels[j * 4 + 0];
    const int lj1 = (int)labels[j * 4 + 1];
    const int lj2 = (int)labels[j * 4 + 2];
    const int lj3 = (int)labels[j * 4 + 3];

#pragma unroll
    for (int r = 0; r < 8; ++r) {
      const float s  = acc[r];
      const float ls = __logf(s + kEPS);
      const float ls2 = ls * ls;
      const int t = trow[r];
      const int i = ibase + r + 8 * hf;
      const bool e0 = (t == lj0), e1 = (t == lj1), e2 = (t == lj2), e3 = (t == lj3);
      const float pf = (e0 && (i != j)) ? 1.f : 0.f;
      const float n0 = e3 ? 0.f : 1.f;               // ~eq3
      const float n1 = (e3 && !e2) ? 1.f : 0.f;      // eq3 & ~eq2
      const float n2 = (e2 && !e1) ? 1.f : 0.f;      // eq2 & ~eq1
      const float n3 = (e1 && !e0) ? 1.f : 0.f;      // eq1 & ~eq0
      const float w0 = n0 + n1 + n2 + n3;
      const float w1 = a0 * n0 + a1 * n1 + a2 * n2 + a3 * n3;
      const float w2 = a0q * n0 + a1q * n1 + a2q * n2 + a3q * n3;
      accq[r][0] += pf;
      accq[r][1] += pf * ls;
      accq[r][2] += pf * ls2;
      accq[r][3] += w0;
      accq[r][4] += w1;
      accq[r][5] += w2;
      accq[r][6] += w0 * ls;
      accq[r][7] += w1 * ls;
      accq[r][8] += w0 * ls2;
    }
  }

  // reduce each accumulator across the 16 lanes of this half (cols of the tile)
#pragma unroll
  for (int r = 0; r < 8; ++r) {
#pragma unroll
    for (int q = 0; q < 9; ++q) {
      float v = accq[r][q];
      v += __shfl_xor(v, 1, 32);
      v += __shfl_xor(v, 2, 32);
      v += __shfl_xor(v, 4, 32);
      v += __shfl_xor(v, 8, 32);
      accq[r][q] = v;
    }
  }
  if (l15 == 0) {
    const int i0 = ibase + 8 * hf;
#pragma unroll
    for (int r = 0; r < 8; ++r)
#pragma unroll
      for (int q = 0; q < 9; ++q)
        atomicAdd(&agg[(i0 + r) * 9 + q], accq[r][q]);
  }
}

__global__ __launch_bounds__(256) void lr_phase2(
    const float* __restrict__ agg, float* __restrict__ out)
{
  __shared__ float red[256];
  const int i = blockIdx.x * 256 + threadIdx.x;
  const float* a = agg + i * 9;
  const float c  = a[0], S1 = a[1], S2 = a[2];
  const float P0 = a[3], P1 = a[4], P2 = a[5];
  const float Q0 = a[6], Q1 = a[7], R0 = a[8];
  const float li = S2 * P0 - 2.f * S1 * (Q0 + P1) + c * (R0 + 2.f * Q1 + P2);
  red[threadIdx.x] = li;
  __syncthreads();
  for (int s = 128; s > 0; s >>= 1) {
    if (threadIdx.x < s) red[threadIdx.x] += red[threadIdx.x + s];
    __syncthreads();
  }
  if (threadIdx.x == 0) atomicAdd(out, red[0]);
}

extern "C" void kernel_launch(void* const* d_in, const int* in_sizes, int n_in,
                              void* d_out, int out_size, void* d_ws, size_t ws_size,
                              hipStream_t stream) {
  const float* x = (const float*)d_in[0];               // 4096x128 f32
  const long long* labels = (const long long*)d_in[1];  // 4096x4 int64
  float* out = (float*)d_out;                           // scalar f32
  float* agg = (float*)d_ws;                            // 4096*9 f32 row aggregates

  hipMemsetAsync(agg, 0, kN * 9 * sizeof(float), stream);
  hipMemsetAsync(out, 0, sizeof(float), stream);

  // a_m = 0.1 * ( log(OMEGA+EPS) - log(OMEGA^(kk-m+1) + EPS) ), m=0..3 -> exps 5,4,3,2
  float a[4];
  for (int m = 0; m < 4; ++m) {
    const float lp = logf(0.1f + 1e-6f) - logf(powf(0.1f, (float)(5 - m)) + 1e-6f);
    a[m] = 0.1f * lp;
  }

  dim3 grid(kN / 64, 8);
  lr_phase1<<<grid, dim3(256), 0, stream>>>(x, labels, agg, a[0], a[1], a[2], a[3]);
  lr_phase2<<<kN / 256, dim3(256), 0, stream>>>(agg, out);
}